// PhaseQuantizer_50122268345058
// MI455X (gfx1250) — compile-verified
//
#include <hip/hip_runtime.h>
#include <cstdint>
#include <math.h>

#define PI_F       3.14159265358979323846f
#define TWO_PI_F   6.28318530717958647692f
#define HALF_PI_F  1.57079632679489661923f
#define LOG2E_F    1.4426950408889634f

typedef __attribute__((ext_vector_type(2))) float v2f;
typedef __attribute__((ext_vector_type(8))) float v8f;

// ---------------------------------------------------------------------------
// Soft phase quantization of one complex element.
// kexp = log2(e) / (temp + 1e-6)
// centers = {0, +pi/2, pi, -pi/2}; delta_k = wrap(phase - c_k) in (-pi, pi].
// phase in [-pi,pi], so wrapping is one conditional +-2pi; |delta_2| = pi-|phase|.
// ---------------------------------------------------------------------------
__device__ __forceinline__ void quant_step(float re, float im, float s_re, float s_im,
                                           float kexp, float& q_re, float& q_im) {
  float phi = atan2f(im, re);
  float d0 = phi;
  float d2 = PI_F - fabsf(phi);
  float d1 = phi - HALF_PI_F; d1 = (d1 < -PI_F) ? (d1 + TWO_PI_F) : d1;
  float d3 = phi + HALF_PI_F; d3 = (d3 >  PI_F) ? (d3 - TWO_PI_F) : d3;
  float l0 = d0 * d0, l1 = d1 * d1, l2 = d2 * d2, l3 = d3 * d3;
  float mn = fminf(fminf(l0, l1), fminf(l2, l3));   // softmax stabilization
  float e0 = __builtin_amdgcn_exp2f((mn - l0) * kexp);
  float e1 = __builtin_amdgcn_exp2f((mn - l1) * kexp);
  float e2 = __builtin_amdgcn_exp2f((mn - l2) * kexp);
  float e3 = __builtin_amdgcn_exp2f((mn - l3) * kexp);
  float inv = 1.0f / (e0 + e1 + e2 + e3);
  q_re = (e0 - e2) * inv * s_re;
  q_im = (e1 - e3) * inv * s_im;
}

// Fixed-order wave32 butterfly reduction (deterministic).
__device__ __forceinline__ float wave_sum(float v) {
  v += __shfl_xor(v, 16);
  v += __shfl_xor(v, 8);
  v += __shfl_xor(v, 4);
  v += __shfl_xor(v, 2);
  v += __shfl_xor(v, 1);
  return v;
}

// Reduce 4 per-thread accumulators across a 256-thread block (8 wave32s)
// and write one float4 partial per block. Fixed order -> deterministic.
__device__ __forceinline__ void block_write_partials(float sr, float si, float cr, float ci,
                                                     float* xred /*shared, >=32 floats*/,
                                                     float* __restrict__ part) {
  sr = wave_sum(sr); si = wave_sum(si); cr = wave_sum(cr); ci = wave_sum(ci);
  const int lane = threadIdx.x & 31;
  const int wv   = threadIdx.x >> 5;
  if (lane == 0) {
    xred[wv * 4 + 0] = sr; xred[wv * 4 + 1] = si;
    xred[wv * 4 + 2] = cr; xred[wv * 4 + 3] = ci;
  }
  __syncthreads();
  if (threadIdx.x == 0) {
    float a = 0.f, b = 0.f, c = 0.f, d = 0.f;
    const int nw = blockDim.x >> 5;
    for (int w = 0; w < nw; ++w) {
      a += xred[w * 4 + 0]; b += xred[w * 4 + 1];
      c += xred[w * 4 + 2]; d += xred[w * 4 + 3];
    }
    part[blockIdx.x * 4 + 0] = a;
    part[blockIdx.x * 4 + 1] = b;
    part[blockIdx.x * 4 + 2] = c;
    part[blockIdx.x * 4 + 3] = d;
  }
}

#define PQ_ACC(RE, IM) do {                       \
    float ar_ = fabsf(RE), ai_ = fabsf(IM);       \
    bool br_ = ar_ > ai_;                         \
    sr += br_ ? ar_ : 0.f; cr += br_ ? 1.f : 0.f; \
    si += br_ ? 0.f : ai_; ci += br_ ? 0.f : 1.f; \
  } while (0)

#define PQ_ISSUE_ASYNC(LRE, LIM, IDX) do {                                     \
    unsigned long long ga_ = (unsigned long long)(uintptr_t)(re4 + (IDX));     \
    unsigned long long gb_ = (unsigned long long)(uintptr_t)(im4 + (IDX));     \
    asm volatile("global_load_async_to_lds_b128 %0, %1, off"                   \
                 :: "v"(LRE), "v"(ga_) : "memory");                            \
    asm volatile("global_load_async_to_lds_b128 %0, %1, off"                   \
                 :: "v"(LIM), "v"(gb_) : "memory");                            \
  } while (0)

// ---------------------------------------------------------------------------
// Kernel 1: masked-abs stats of (w_re, w_im). Quadrant mask == |re| vs |im|
// (no atan2 needed). Streams through LDS with double-buffered CDNA5 async
// global->LDS copies. Trip count `kfull` is block-uniform (grid stride >=
// block span), so the ASYNCcnt wait immediates are exact at wave level.
// Each lane owns its own 16B LDS slot -> only ASYNCcnt/DScnt ordering needed.
// ---------------------------------------------------------------------------
__global__ void pq_stats1(const float* __restrict__ re, const float* __restrict__ im,
                          float* __restrict__ part, int n, int n4) {
  __shared__ float4 bre[2][256];
  __shared__ float4 bim[2][256];
  __shared__ float  xred[32];
  const int t = threadIdx.x;
  const unsigned lre0 = (unsigned)(uintptr_t)(&bre[0][t]);  // LDS byte offsets
  const unsigned lre1 = (unsigned)(uintptr_t)(&bre[1][t]);
  const unsigned lim0 = (unsigned)(uintptr_t)(&bim[0][t]);
  const unsigned lim1 = (unsigned)(uintptr_t)(&bim[1][t]);
  const float4* re4 = (const float4*)re;
  const float4* im4 = (const float4*)im;

  float sr = 0.f, si = 0.f, cr = 0.f, ci = 0.f;
  const int i0     = blockIdx.x * blockDim.x + t;
  const int stride = gridDim.x * blockDim.x;

  // iterations guaranteed valid for EVERY thread in this block
  const int maxbase = blockIdx.x * blockDim.x + (blockDim.x - 1);
  const int kfull   = (n4 > maxbase) ? ((n4 - 1 - maxbase) / stride + 1) : 0;

  if (kfull > 0) {
    PQ_ISSUE_ASYNC(lre0, lim0, i0);                 // prologue -> buffer 0
  }
  for (int k = 0; k < kfull; ++k) {
    const int cur = k & 1;
    if (k + 1 < kfull) {
      // WAR: buffer cur^1's ds_loads (iteration k-1) must land in VGPRs
      // before the async engine overwrites it.
      asm volatile("s_wait_dscnt 0" ::: "memory");
      PQ_ISSUE_ASYNC(cur ? lre0 : lre1, cur ? lim0 : lim1, i0 + (k + 1) * stride);
      asm volatile("s_wait_asynccnt 2" ::: "memory");  // current buffer's 2 loads done
    } else {
      asm volatile("s_wait_asynccnt 0" ::: "memory");
    }
    float4 r = bre[cur][t];
    float4 m = bim[cur][t];
    PQ_ACC(r.x, m.x); PQ_ACC(r.y, m.y); PQ_ACC(r.z, m.z); PQ_ACC(r.w, m.w);
  }
  // ragged last iteration (at most one extra per thread): direct loads
  const int itail = i0 + kfull * stride;
  if (itail < n4) {
    float4 r = re4[itail];
    float4 m = im4[itail];
    PQ_ACC(r.x, m.x); PQ_ACC(r.y, m.y); PQ_ACC(r.z, m.z); PQ_ACC(r.w, m.w);
  }
  // scalar tail (n not multiple of 4)
  const int rem = n - 4 * n4;
  if (blockIdx.x == 0 && t < rem) {
    PQ_ACC(re[4 * n4 + t], im[4 * n4 + t]);
  }
  block_write_partials(sr, si, cr, ci, xred, part);
}

// ---------------------------------------------------------------------------
// Finalize: single block. Per-thread strided f32 accumulation, then the
// cross-lane reduction of wave 0 is folded through V_WMMA_F32_16X16X4_F32:
//   B vgpr0/vgpr1 carry two per-lane quantities (B rows {0,2}/{1,3});
//   A is a 0/1 selector (row0 -> K{0,2}, row1 -> K{1,3}), so
//   D[0][n] = a_n + a_{n+16},  D[1][n] = b_n + b_{n+16}.
// One WMMA folds 32 lanes -> 16 for two quantities; 4 shfl_xor steps finish.
// Fixed order -> deterministic.
// ---------------------------------------------------------------------------
__global__ void pq_finalize(const float* __restrict__ part, int nb, float* __restrict__ sout) {
  __shared__ float acc[4 * 256];
  const int t = threadIdx.x;
  float a = 0.f, b = 0.f, c = 0.f, d = 0.f;
  for (int i = t; i < nb; i += blockDim.x) {
    a += part[4 * i + 0]; b += part[4 * i + 1];
    c += part[4 * i + 2]; d += part[4 * i + 3];
  }
  acc[t] = a; acc[256 + t] = b; acc[512 + t] = c; acc[768 + t] = d;
  __syncthreads();
  if (t < 32) {                                // wave 0, full EXEC (WMMA requirement)
    float ra = 0.f, rb = 0.f, rc = 0.f, rd = 0.f;
    for (int j = t; j < 256; j += 32) {
      ra += acc[j]; rb += acc[256 + j]; rc += acc[512 + j]; rd += acc[768 + j];
    }
    v2f A;  A.x  = (t == 0 || t == 16) ? 1.f : 0.f;   // A[0][0], A[0][2]
            A.y  = (t == 1 || t == 17) ? 1.f : 0.f;   // A[1][1], A[1][3]
    v2f B1; B1.x = ra; B1.y = rb;
    v2f B2; B2.x = rc; B2.y = rd;
    v8f z = {0.f, 0.f, 0.f, 0.f, 0.f, 0.f, 0.f, 0.f};
    v8f D1 = __builtin_amdgcn_wmma_f32_16x16x4_f32(false, A, false, B1, (short)0, z, false, false);
    v8f D2 = __builtin_amdgcn_wmma_f32_16x16x4_f32(false, A, false, B2, (short)0, z, false, false);
    float fa = D1[0];   // lanes 0..15: ra_n + ra_{n+16}
    float fb = D1[1];   // lanes 0..15: rb_n + rb_{n+16}
    float fc = D2[0];
    float fd = D2[1];
    for (int mask = 8; mask > 0; mask >>= 1) {
      fa += __shfl_xor(fa, mask);
      fb += __shfl_xor(fb, mask);
      fc += __shfl_xor(fc, mask);
      fd += __shfl_xor(fd, mask);
    }
    if (t == 0) {
      float s_re = (fc > 0.f) ? (fa / fmaxf(fc, 1.f)) : 0.f;
      float s_im = (fd > 0.f) ? (fb / fmaxf(fd, 1.f)) : 0.f;
      sout[0] = fmaxf(s_re, 1e-6f);
      sout[1] = fmaxf(s_im, 1e-6f);
    }
  }
}

// ---------------------------------------------------------------------------
// Kernel 2: recompute q1 on the fly, reduce masked-abs stats of err1 = w - q1.
// ---------------------------------------------------------------------------
__global__ void pq_stats2(const float* __restrict__ re, const float* __restrict__ im,
                          const float* __restrict__ tptr, const float* __restrict__ svals,
                          float* __restrict__ part, int n, int n4) {
  __shared__ float xred[32];
  const float s_re1 = svals[0], s_im1 = svals[1];
  const float kexp  = LOG2E_F / (tptr[0] + 1e-6f);
  const float4* re4 = (const float4*)re;
  const float4* im4 = (const float4*)im;
  float sr = 0.f, si = 0.f, cr = 0.f, ci = 0.f;
  const int stride = gridDim.x * blockDim.x;

#define PQ_STEP2(RE, IM) do {                                   \
    float qr_, qi_;                                             \
    quant_step(RE, IM, s_re1, s_im1, kexp, qr_, qi_);           \
    float er_ = (RE) - qr_, ei_ = (IM) - qi_;                   \
    PQ_ACC(er_, ei_);                                           \
  } while (0)

  for (int i = blockIdx.x * blockDim.x + threadIdx.x; i < n4; i += stride) {
    float4 r = re4[i];
    float4 m = im4[i];
    PQ_STEP2(r.x, m.x); PQ_STEP2(r.y, m.y); PQ_STEP2(r.z, m.z); PQ_STEP2(r.w, m.w);
  }
  const int rem = n - 4 * n4;
  if (blockIdx.x == 0 && threadIdx.x < rem) {
    PQ_STEP2(re[4 * n4 + threadIdx.x], im[4 * n4 + threadIdx.x]);
  }
  block_write_partials(sr, si, cr, ci, xred, part);
}

// ---------------------------------------------------------------------------
// Kernel 3: qw = q1 + q2 (recompute q1, then q2 from err1). float4 stores.
// ---------------------------------------------------------------------------
__global__ void pq_output(const float* __restrict__ re, const float* __restrict__ im,
                          const float* __restrict__ tptr, const float* __restrict__ svals,
                          float* __restrict__ out_re, float* __restrict__ out_im,
                          int n, int n4) {
  const float s_re1 = svals[0], s_im1 = svals[1];
  const float s_re2 = svals[2], s_im2 = svals[3];
  const float kexp  = LOG2E_F / (tptr[0] + 1e-6f);
  const float4* re4 = (const float4*)re;
  const float4* im4 = (const float4*)im;
  float4* or4 = (float4*)out_re;
  float4* oi4 = (float4*)out_im;
  const int stride = gridDim.x * blockDim.x;

#define PQ_OUT(RE, IM, OR, OI) do {                             \
    float q1r_, q1i_, q2r_, q2i_;                               \
    quant_step(RE, IM, s_re1, s_im1, kexp, q1r_, q1i_);         \
    float er_ = (RE) - q1r_, ei_ = (IM) - q1i_;                 \
    quant_step(er_, ei_, s_re2, s_im2, kexp, q2r_, q2i_);       \
    OR = q1r_ + q2r_;                                           \
    OI = q1i_ + q2i_;                                           \
  } while (0)

  for (int i = blockIdx.x * blockDim.x + threadIdx.x; i < n4; i += stride) {
    float4 r = re4[i];
    float4 m = im4[i];
    float4 qr, qi;
    PQ_OUT(r.x, m.x, qr.x, qi.x);
    PQ_OUT(r.y, m.y, qr.y, qi.y);
    PQ_OUT(r.z, m.z, qr.z, qi.z);
    PQ_OUT(r.w, m.w, qr.w, qi.w);
    or4[i] = qr;
    oi4[i] = qi;
  }
  const int rem = n - 4 * n4;
  if (blockIdx.x == 0 && threadIdx.x < rem) {
    const int j = 4 * n4 + threadIdx.x;
    float orr, oii;
    PQ_OUT(re[j], im[j], orr, oii);
    out_re[j] = orr;
    out_im[j] = oii;
  }
}

// ---------------------------------------------------------------------------
// Host launch. ws layout (floats):
//   [0..3]  s_re1, s_im1, s_re2, s_im2
//   [8 ..]  stage-1 partials: nb * 4 floats
//   [...]   stage-2 partials: nb * 4 floats
// ---------------------------------------------------------------------------
extern "C" void kernel_launch(void* const* d_in, const int* in_sizes, int n_in,
                              void* d_out, int out_size, void* d_ws, size_t ws_size,
                              hipStream_t stream) {
  (void)n_in; (void)out_size;
  const float* w_re = (const float*)d_in[0];
  const float* w_im = (const float*)d_in[1];
  const float* temp = (const float*)d_in[2];
  const int n  = in_sizes[0];
  const int n4 = n >> 2;
  float* out_re = (float*)d_out;
  float* out_im = out_re + n;
  float* wsf = (float*)d_ws;

  int nb = 2048;
  const size_t need = (size_t)(8 + 8 * nb) * sizeof(float);
  if (ws_size < need) {
    long avail = (long)(ws_size / sizeof(float)) - 8;
    nb = (int)(avail / 8);
    if (nb < 1) nb = 1;
  }
  float* svals = wsf;
  float* part1 = wsf + 8;
  float* part2 = part1 + 4 * (size_t)nb;

  const dim3 blk(256);
  pq_stats1<<<nb, blk, 0, stream>>>(w_re, w_im, part1, n, n4);
  pq_finalize<<<1, blk, 0, stream>>>(part1, nb, svals);
  pq_stats2<<<nb, blk, 0, stream>>>(w_re, w_im, temp, svals, part2, n, n4);
  pq_finalize<<<1, blk, 0, stream>>>(part2, nb, svals + 2);

  int b3 = (n4 + 255) / 256;
  if (b3 > 8192) b3 = 8192;
  if (b3 < 1) b3 = 1;
  pq_output<<<b3, blk, 0, stream>>>(w_re, w_im, temp, svals, out_re, out_im, n, n4);
}